// RGraphAttention_38482906972415
// MI455X (gfx1250) — compile-verified
//
#include <hip/hip_runtime.h>

#define NNODES 50000
#define NEDGES 800000
#define HID    64
#define NEG_SLOPE 0.01f

// LDS weight layout: pair-interleaved K, padded pitch for bank-conflict-free b64 reads.
// lds[(k/2)*PITCH + 2*n + (k&1)] = W[k][n]; PITCH=160 floats (640B = 128 mod 256
// -> half-wave 1 lands on banks 32..63 while half-wave 0 uses banks 0..31).
#define WPITCH 160
#define WLDS   (32 * WPITCH)

typedef __attribute__((ext_vector_type(2))) float v2f;
typedef __attribute__((ext_vector_type(8))) float v8f;

__device__ __forceinline__ float leaky(float x) { return x >= 0.f ? x : NEG_SLOPE * x; }

// order-preserving float->uint key for atomicMax-based segment max
__device__ __forceinline__ unsigned fkey(float f) {
    unsigned u = __float_as_uint(f);
    return (u & 0x80000000u) ? ~u : (u | 0x80000000u);
}
__device__ __forceinline__ float funkey(unsigned k) {
    unsigned u = (k & 0x80000000u) ? (k & 0x7fffffffu) : ~k;
    return __uint_as_float(u);
}

// ---------------- init workspace accumulators ----------------
__global__ void k_init(unsigned* __restrict__ smax_key, float* __restrict__ denom,
                       float* __restrict__ agg) {
    int i = blockIdx.x * blockDim.x + threadIdx.x;
    if (i < NNODES) { smax_key[i] = 0u; denom[i] = 0.f; }
    if (i < NNODES * HID) agg[i] = 0.f;
}

// ---------------- h = node_feat @ Wn + bn (WMMA f32 16x16x4) ----------------
__global__ __launch_bounds__(256) void k_node_proj(const float* __restrict__ nf,
                                                   const float* __restrict__ Wn,
                                                   const float* __restrict__ bn,
                                                   float* __restrict__ h) {
    __shared__ float ldsW[WLDS];
    for (int i = threadIdx.x; i < HID * HID; i += 256) {
        int k = i >> 6, n = i & 63;
        ldsW[(k >> 1) * WPITCH + 2 * n + (k & 1)] = Wn[i];
    }
    __syncthreads();

    const int wave = threadIdx.x >> 5;
    const int lane = threadIdx.x & 31;
    const int tile = blockIdx.x * 8 + wave;
    if (tile >= NNODES / 16) return;
    const int ln15 = lane & 15;
    const int hf   = lane >> 4;

    const float* arow  = nf + (size_t)(tile * 16 + ln15) * HID;
    const float* bbase = ldsW + hf * WPITCH + 2 * ln15; // + (k0/2)*WPITCH + 32*nt
    v8f c[4] = {};
    for (int k0 = 0; k0 < HID; k0 += 4) {
        v2f a = *(const v2f*)(arow + k0 + 2 * hf);
        #pragma unroll
        for (int nt = 0; nt < 4; nt++) {
            v2f b = *(const v2f*)(bbase + (k0 >> 1) * WPITCH + 32 * nt); // ds_load_b64
            c[nt] = __builtin_amdgcn_wmma_f32_16x16x4_f32(false, a, false, b,
                                                          (short)0, c[nt], false, false);
        }
    }
    #pragma unroll
    for (int nt = 0; nt < 4; nt++) {
        float bias = bn[nt * 16 + ln15];
        #pragma unroll
        for (int r = 0; r < 8; r++)
            h[(size_t)(tile * 16 + r + 8 * hf) * HID + nt * 16 + ln15] = c[nt][r] + bias;
    }
}

// ---------------- fused edge kernel ----------------
__global__ __launch_bounds__(256) void k_edge(const float* __restrict__ ef,
                                              const float* __restrict__ gv,
                                              const float* __restrict__ We,
                                              const float* __restrict__ be,
                                              const float* __restrict__ Wg,
                                              const float* __restrict__ bg,
                                              const float* __restrict__ wea_g, // w_edge_att
                                              const float* __restrict__ wna_g, // w_node_att
                                              const int* __restrict__ src,
                                              const int* __restrict__ dst,
                                              const float* __restrict__ h,
                                              float* __restrict__ e_new_out,
                                              float* __restrict__ score_ws,
                                              unsigned* __restrict__ smax_key) {
    __shared__ float ldsWe[WLDS];
    __shared__ float ldsWg[WLDS];
    for (int i = threadIdx.x; i < HID * HID; i += 256) {
        int k = i >> 6, n = i & 63;
        int o = (k >> 1) * WPITCH + 2 * n + (k & 1);
        ldsWe[o] = We[i];
        ldsWg[o] = Wg[i];
    }
    __syncthreads();

    const int wave = threadIdx.x >> 5;
    const int lane = threadIdx.x & 31;
    const int tile = blockIdx.x * 8 + wave; // 50000 tiles of 16 edges
    const int ln15 = lane & 15;
    const int hf   = lane >> 4;
    const int ebase = tile * 16;

    float wea[4], wna[4], bev[4], bgv[4];
    #pragma unroll
    for (int nt = 0; nt < 4; nt++) {
        wea[nt] = wea_g[nt * 16 + ln15];
        wna[nt] = wna_g[nt * 16 + ln15];
        bev[nt] = be[nt * 16 + ln15];
        bgv[nt] = bg[nt * 16 + ln15];
    }
    int srcm[8], dstm[8];
    #pragma unroll
    for (int r = 0; r < 8; r++) {
        srcm[r] = src[ebase + 8 * hf + r];
        dstm[r] = dst[ebase + 8 * hf + r];
    }
    // warm the L2/L0 lines for the h gathers while the WMMA loop runs
    #pragma unroll
    for (int r = 0; r < 8; r++) {
        __builtin_prefetch(&h[(size_t)srcm[r] * HID + ln15 * 4], 0, 3);
        __builtin_prefetch(&h[(size_t)dstm[r] * HID + ln15 * 4], 0, 3);
    }

    // WMMA projections (K=64 in steps of 4)
    const float* ea = ef + (size_t)(ebase + ln15) * HID;
    const float* ga = gv + (size_t)(ebase + ln15) * HID;
    const float* be1 = ldsWe + hf * WPITCH + 2 * ln15;
    const float* be2 = ldsWg + hf * WPITCH + 2 * ln15;
    v8f ce[4] = {}; // e tile
    v8f cg[4] = {}; // gu tile
    for (int k0 = 0; k0 < HID; k0 += 4) {
        v2f ae = *(const v2f*)(ea + k0 + 2 * hf);
        v2f ag = *(const v2f*)(ga + k0 + 2 * hf);
        #pragma unroll
        for (int nt = 0; nt < 4; nt++) {
            v2f b1 = *(const v2f*)(be1 + (k0 >> 1) * WPITCH + 32 * nt);
            v2f b2 = *(const v2f*)(be2 + (k0 >> 1) * WPITCH + 32 * nt);
            ce[nt] = __builtin_amdgcn_wmma_f32_16x16x4_f32(false, ae, false, b1,
                                                           (short)0, ce[nt], false, false);
            cg[nt] = __builtin_amdgcn_wmma_f32_16x16x4_f32(false, ag, false, b2,
                                                           (short)0, cg[nt], false, false);
        }
    }
    #pragma unroll
    for (int nt = 0; nt < 4; nt++)
        #pragma unroll
        for (int r = 0; r < 8; r++) { ce[nt][r] += bev[nt]; cg[nt][r] += bgv[nt]; }

    // gathers (h is 12.8 MB -> L2 resident, prefetched above)
    v8f hs[4], hd[4];
    #pragma unroll
    for (int nt = 0; nt < 4; nt++)
        #pragma unroll
        for (int r = 0; r < 8; r++) {
            hs[nt][r] = h[(size_t)srcm[r] * HID + nt * 16 + ln15];
            hd[nt][r] = h[(size_t)dstm[r] * HID + nt * 16 + ln15];
        }

    // two-way attention logits, reduced across the 16-lane half (wave32 shuffles)
    float al1[8], al2[8];
    #pragma unroll
    for (int r = 0; r < 8; r++) {
        float p1 = 0.f, p2 = 0.f;
        #pragma unroll
        for (int nt = 0; nt < 4; nt++) {
            float eg = ce[nt][r] + cg[nt][r];
            p1 += (hs[nt][r] + eg) * wea[nt];
            p2 += (hd[nt][r] + eg) * wea[nt];
        }
        #pragma unroll
        for (int m = 1; m < 16; m <<= 1) {
            p1 += __shfl_xor(p1, m, 32);
            p2 += __shfl_xor(p2, m, 32);
        }
        float a1 = leaky(p1), a2 = leaky(p2);
        float mx = fmaxf(a1, a2);
        float e1 = __expf(a1 - mx), e2 = __expf(a2 - mx);
        float d = e1 + e2;
        al1[r] = e1 / d;
        al2[r] = e2 / d;
    }

    // e_new (second output, streamed straight to d_out) + score partials
    float psc[8];
    #pragma unroll
    for (int r = 0; r < 8; r++) psc[r] = 0.f;
    #pragma unroll
    for (int nt = 0; nt < 4; nt++) {
        #pragma unroll
        for (int r = 0; r < 8; r++) {
            float en = al1[r] * hs[nt][r] + al2[r] * hd[nt][r] + ce[nt][r];
            e_new_out[(size_t)(ebase + r + 8 * hf) * HID + nt * 16 + ln15] = en;
            psc[r] += (hs[nt][r] + hd[nt][r] + en + cg[nt][r]) * wna[nt];
        }
    }
    #pragma unroll
    for (int r = 0; r < 8; r++) {
        float p = psc[r];
        #pragma unroll
        for (int m = 1; m < 16; m <<= 1) p += __shfl_xor(p, m, 32);
        float sc = leaky(p);
        if (ln15 == r) { // one writer lane per edge
            int e = ebase + 8 * hf + r;
            score_ws[e] = sc;
            atomicMax(&smax_key[dstm[r]], fkey(sc));
        }
    }
}

// ---------------- ex = exp(score - smax[dst]); denom[dst] += ex ----------------
__global__ void k_exden(const float* __restrict__ score, const int* __restrict__ dst,
                        const unsigned* __restrict__ smax_key,
                        float* __restrict__ ex, float* __restrict__ denom) {
    int e = blockIdx.x * blockDim.x + threadIdx.x;
    if (e >= NEDGES) return;
    int d = dst[e];
    float v = __expf(score[e] - funkey(smax_key[d]));
    ex[e] = v;
    atomicAdd(&denom[d], v);
}

// ---------------- agg[dst] += beta * h[src] ----------------
__global__ void k_agg(const float* __restrict__ ex, const float* __restrict__ denom,
                      const int* __restrict__ src, const int* __restrict__ dst,
                      const float* __restrict__ h, float* __restrict__ agg) {
    long long t = (long long)blockIdx.x * blockDim.x + threadIdx.x;
    if (t >= (long long)NEDGES * HID) return;
    int e = (int)(t >> 6);
    int c = (int)(t & 63);
    int d = dst[e];
    float beta = ex[e] / denom[d];
    atomicAdd(&agg[(size_t)d * HID + c], beta * h[(size_t)src[e] * HID + c]);
}

// ---------------- node_out = agg + h ----------------
__global__ void k_nodeout(const float* __restrict__ agg, const float* __restrict__ h,
                          float* __restrict__ node_out) {
    int i = blockIdx.x * blockDim.x + threadIdx.x;
    if (i < NNODES * HID) node_out[i] = agg[i] + h[i];
}

extern "C" void kernel_launch(void* const* d_in, const int* in_sizes, int n_in,
                              void* d_out, int out_size, void* d_ws, size_t ws_size,
                              hipStream_t stream) {
    // inputs (setup_inputs order)
    const float* node_feat = (const float*)d_in[0];
    const float* edge_feat = (const float*)d_in[1];
    const float* guild_vec = (const float*)d_in[2];
    const float* Wn        = (const float*)d_in[3];
    const float* bn        = (const float*)d_in[4];
    const float* We        = (const float*)d_in[5];
    const float* be        = (const float*)d_in[6];
    const float* Wg        = (const float*)d_in[7];
    const float* bg        = (const float*)d_in[8];
    const float* w_node_att = (const float*)d_in[9];
    const float* w_edge_att = (const float*)d_in[10];
    const int*   src       = (const int*)d_in[11];
    const int*   dst       = (const int*)d_in[12];

    // outputs: (node_out [N,H], e_new [E,H]) concatenated
    float* node_out = (float*)d_out;
    float* e_new    = node_out + (size_t)NNODES * HID;

    // workspace layout
    float*    h        = (float*)d_ws;                       // N*H
    float*    score    = h + (size_t)NNODES * HID;           // E
    float*    ex       = score + NEDGES;                     // E
    float*    denom    = ex + NEDGES;                        // N
    unsigned* smax_key = (unsigned*)(denom + NNODES);        // N
    float*    agg      = (float*)(smax_key + NNODES);        // N*H

    hipLaunchKernelGGL(k_init, dim3((NNODES * HID + 255) / 256), dim3(256), 0, stream,
                       smax_key, denom, agg);
    hipLaunchKernelGGL(k_node_proj, dim3((NNODES / 16 + 7) / 8), dim3(256), 0, stream,
                       node_feat, Wn, bn, h);
    hipLaunchKernelGGL(k_edge, dim3(NEDGES / 16 / 8), dim3(256), 0, stream,
                       edge_feat, guild_vec, We, be, Wg, bg,
                       w_edge_att, w_node_att, src, dst, h,
                       e_new, score, smax_key);
    hipLaunchKernelGGL(k_exden, dim3((NEDGES + 255) / 256), dim3(256), 0, stream,
                       score, dst, smax_key, ex, denom);
    hipLaunchKernelGGL(k_agg, dim3((int)(((long long)NEDGES * HID + 255) / 256)), dim3(256), 0,
                       stream, ex, denom, src, dst, h, agg);
    hipLaunchKernelGGL(k_nodeout, dim3((NNODES * HID + 255) / 256), dim3(256), 0, stream,
                       agg, h, node_out);
}